// Joint_model_Soft_Interaction_36558761624264
// MI455X (gfx1250) — compile-verified
//
#include <hip/hip_runtime.h>

#define DD 256
#define SS 4096
#define RR 64
#define BB 16

typedef __attribute__((ext_vector_type(2))) float v2f;
typedef __attribute__((ext_vector_type(8))) float v8f;

// ---------------------------------------------------------------------------
// K1: fold Wk@ws_k -> u, and wg_o+wg_p -> wsum, for both passes.
// ---------------------------------------------------------------------------
__global__ void prep_vectors(const float* __restrict__ Wk1, const float* __restrict__ wsk1,
                             const float* __restrict__ wgo1, const float* __restrict__ wgp1,
                             const float* __restrict__ Wk2, const float* __restrict__ wsk2,
                             const float* __restrict__ wgo2, const float* __restrict__ wgp2,
                             float* __restrict__ u1, float* __restrict__ u2,
                             float* __restrict__ wsum1, float* __restrict__ wsum2) {
    const int t = threadIdx.x;
    if (blockIdx.x == 0) {
        float s = 0.0f;
        for (int j = 0; j < DD; ++j) s += Wk1[t * DD + j] * wsk1[j];
        u1[t] = s;
        wsum1[t] = wgo1[t] + wgp1[t];
    } else {
        float s = 0.0f;
        for (int j = 0; j < DD; ++j) s += Wk2[t * DD + j] * wsk2[j];
        u2[t] = s;
        wsum2[t] = wgo2[t] + wgp2[t];
    }
}

// ---------------------------------------------------------------------------
// K2: pass-1 attention over the 64 p-rows per batch: scores, softmax, pbar.
// ---------------------------------------------------------------------------
__global__ void pass1_attn(const float* __restrict__ p, const float* __restrict__ u1,
                           float* __restrict__ pbar) {
    __shared__ float w[RR];
    const int b = blockIdx.x, t = threadIdx.x;
    if (t < RR) {
        const float* row = p + (size_t)(b * RR + t) * DD;
        float sk = 0.0f;
        for (int j = 0; j < DD; ++j) sk += row[j] * u1[j];
        w[t] = sk;
    }
    __syncthreads();
    if (t == 0) {
        float mx = -3.0e38f;
        for (int l = 0; l < RR; ++l) mx = fmaxf(mx, w[l]);
        float s = 0.0f;
        for (int l = 0; l < RR; ++l) { float e = __expf(w[l] - mx); w[l] = e; s += e; }
        const float inv = 1.0f / s;
        for (int l = 0; l < RR; ++l) w[l] *= inv;
    }
    __syncthreads();
    float acc = 0.0f;
    for (int l = 0; l < RR; ++l) acc += w[l] * p[(size_t)(b * RR + l) * DD + t];
    pbar[b * DD + t] = acc;
}

// ---------------------------------------------------------------------------
// K3/K8: C(16x256) = A(16x256) @ W(256x256) + bias, via v_wmma_f32_16x16x4_f32.
// One wave per 16-wide N tile; 64 K-steps of 16x16x4.
// ---------------------------------------------------------------------------
__global__ void wmma_gemm16(const float* __restrict__ A, const float* __restrict__ W,
                            const float* __restrict__ bias, float* __restrict__ C) {
    const int lane = threadIdx.x & 31;
    const int half = lane >> 4;        // 0: K pair {0,1}, 1: K pair {2,3}
    const int l16  = lane & 15;
    const int n    = blockIdx.x * 16 + l16;
    v8f acc = {};
    for (int k0 = 0; k0 < DD; k0 += 4) {
        const int ka = k0 + half * 2;
        v2f a, b;
        a.x = A[l16 * DD + ka];
        a.y = A[l16 * DD + ka + 1];
        b.x = W[(size_t)(ka)     * DD + n];
        b.y = W[(size_t)(ka + 1) * DD + n];
        acc = __builtin_amdgcn_wmma_f32_16x16x4_f32(false, a, false, b,
                                                    (short)0, acc, false, false);
    }
    const float bv = bias[n];
#pragma unroll
    for (int j = 0; j < 8; ++j) {
        const int m = half * 8 + j;    // D layout: VGPR j -> M=j (lanes 0-15) / M=8+j
        C[m * DD + n] = acc[j] + bv;
    }
}

// ---------------------------------------------------------------------------
// K4/K9: gbias[b] = ctx[b]·wg_o + bg
// ---------------------------------------------------------------------------
__global__ void gate_bias(const float* __restrict__ ctx, const float* __restrict__ wg_o,
                          const float* __restrict__ bg, float* __restrict__ gbias) {
    __shared__ float red[DD];
    const int b = blockIdx.x, t = threadIdx.x;
    red[t] = ctx[b * DD + t] * wg_o[t];
    __syncthreads();
    for (int s = 128; s > 0; s >>= 1) {
        if (t < s) red[t] += red[t + s];
        __syncthreads();
    }
    if (t == 0) gbias[b] = red[0] + bg[0];
}

// ---------------------------------------------------------------------------
// K5: streaming fused pass over x: gate + x_out, plus masked pass-2 scores.
// 256 threads = 8 waves, one wave per row (256 floats, two float4 per lane).
// ---------------------------------------------------------------------------
__global__ void fused_x_kernel(const float* __restrict__ x, const float* __restrict__ ctx1,
                               const float* __restrict__ wsum1, const float* __restrict__ u2,
                               const float* __restrict__ gbias1, const int* __restrict__ mask,
                               float* __restrict__ xout, float* __restrict__ sk2) {
    __shared__ float sctx[DD], swsum[DD], su2[DD];
    const int t = threadIdx.x;
    const int rowBase = blockIdx.x * 8;
    const int b = rowBase >> 12;                 // 4096 rows per batch
    sctx[t] = ctx1[b * DD + t];
    swsum[t] = wsum1[t];
    su2[t] = u2[t];
    __syncthreads();
    const int lane = t & 31;
    const int row = rowBase + (t >> 5);
    const float* xr = x + (size_t)row * DD;
    const int da = lane * 4, db = 128 + lane * 4;
    const float4 xa = *(const float4*)(xr + da);
    const float4 xb = *(const float4*)(xr + db);
    float dot = xa.x * swsum[da] + xa.y * swsum[da + 1] + xa.z * swsum[da + 2] + xa.w * swsum[da + 3]
              + xb.x * swsum[db] + xb.y * swsum[db + 1] + xb.z * swsum[db + 2] + xb.w * swsum[db + 3];
#pragma unroll
    for (int m = 16; m > 0; m >>= 1) dot += __shfl_xor(dot, m, 32);
    const float g = 1.0f / (1.0f + __expf(-(dot + gbias1[b])));
    float4 oa, ob;
    oa.x = g * sctx[da]     + 2.0f * xa.x;
    oa.y = g * sctx[da + 1] + 2.0f * xa.y;
    oa.z = g * sctx[da + 2] + 2.0f * xa.z;
    oa.w = g * sctx[da + 3] + 2.0f * xa.w;
    ob.x = g * sctx[db]     + 2.0f * xb.x;
    ob.y = g * sctx[db + 1] + 2.0f * xb.y;
    ob.z = g * sctx[db + 2] + 2.0f * xb.z;
    ob.w = g * sctx[db + 3] + 2.0f * xb.w;
    float dot2 = oa.x * su2[da] + oa.y * su2[da + 1] + oa.z * su2[da + 2] + oa.w * su2[da + 3]
               + ob.x * su2[db] + ob.y * su2[db + 1] + ob.z * su2[db + 2] + ob.w * su2[db + 3];
#pragma unroll
    for (int m = 16; m > 0; m >>= 1) dot2 += __shfl_xor(dot2, m, 32);
    float* orow = xout + (size_t)row * DD;
    *(float4*)(orow + da) = oa;
    *(float4*)(orow + db) = ob;
    if (lane == 0) {
        const int l = row & (SS - 1);
        sk2[b * SS + l] = (mask[b * SS + l] == 0) ? -1.0e9f : dot2;
    }
}

// ---------------------------------------------------------------------------
// K6: softmax statistics (max, sum-of-exp) over 4096 scores per batch.
// ---------------------------------------------------------------------------
__global__ void softmax_stats(const float* __restrict__ sk, float* __restrict__ smax,
                              float* __restrict__ ssum) {
    __shared__ float red[DD];
    const int b = blockIdx.x, t = threadIdx.x;
    const float* row = sk + b * SS;
    float mx = -3.0e38f;
    for (int l = t; l < SS; l += DD) mx = fmaxf(mx, row[l]);
    red[t] = mx;
    __syncthreads();
    for (int s = 128; s > 0; s >>= 1) { if (t < s) red[t] = fmaxf(red[t], red[t + s]); __syncthreads(); }
    mx = red[0];
    __syncthreads();
    float sm = 0.0f;
    for (int l = t; l < SS; l += DD) sm += __expf(row[l] - mx);
    red[t] = sm;
    __syncthreads();
    for (int s = 128; s > 0; s >>= 1) { if (t < s) red[t] += red[t + s]; __syncthreads(); }
    if (t == 0) { smax[b] = mx; ssum[b] = red[0]; }
}

// ---------------------------------------------------------------------------
// K7: xbar[b,:] = sum_l softmax(sk)[l] * x_out[b,l,:]   (coalesced over d)
// ---------------------------------------------------------------------------
__global__ void xbar_kernel(const float* __restrict__ xo, const float* __restrict__ sk,
                            const float* __restrict__ smax, const float* __restrict__ ssum,
                            float* __restrict__ xbar) {
    __shared__ float w[DD];
    const int b = blockIdx.x, d = threadIdx.x;
    const float mx = smax[b], inv = 1.0f / ssum[b];
    const float* base = xo + (size_t)b * SS * DD;
    const float* skr = sk + b * SS;
    float acc = 0.0f;
    for (int l0 = 0; l0 < SS; l0 += DD) {
        __syncthreads();
        w[d] = __expf(skr[l0 + d] - mx) * inv;
        __syncthreads();
        for (int l = 0; l < DD; ++l) acc += w[l] * base[(size_t)(l0 + l) * DD + d];
    }
    xbar[b * DD + d] = acc;
}

// ---------------------------------------------------------------------------
// K10: p_out rows: g = sigmoid(gbias2[b] + p·wsum2); p_out = g*ctx2[b] + 2*p
// ---------------------------------------------------------------------------
__global__ void fused_p_kernel(const float* __restrict__ p, const float* __restrict__ ctx2,
                               const float* __restrict__ wsum2, const float* __restrict__ gbias2,
                               float* __restrict__ pout) {
    __shared__ float sctx[DD], swsum[DD];
    const int t = threadIdx.x;
    const int rowBase = blockIdx.x * 8;
    const int b = rowBase >> 6;                  // 64 rows per batch
    sctx[t] = ctx2[b * DD + t];
    swsum[t] = wsum2[t];
    __syncthreads();
    const int lane = t & 31;
    const int row = rowBase + (t >> 5);
    const float* pr = p + (size_t)row * DD;
    const int da = lane * 4, db = 128 + lane * 4;
    const float4 xa = *(const float4*)(pr + da);
    const float4 xb = *(const float4*)(pr + db);
    float dot = xa.x * swsum[da] + xa.y * swsum[da + 1] + xa.z * swsum[da + 2] + xa.w * swsum[da + 3]
              + xb.x * swsum[db] + xb.y * swsum[db + 1] + xb.z * swsum[db + 2] + xb.w * swsum[db + 3];
#pragma unroll
    for (int m = 16; m > 0; m >>= 1) dot += __shfl_xor(dot, m, 32);
    const float g = 1.0f / (1.0f + __expf(-(dot + gbias2[b])));
    float4 oa, ob;
    oa.x = g * sctx[da]     + 2.0f * xa.x;
    oa.y = g * sctx[da + 1] + 2.0f * xa.y;
    oa.z = g * sctx[da + 2] + 2.0f * xa.z;
    oa.w = g * sctx[da + 3] + 2.0f * xa.w;
    ob.x = g * sctx[db]     + 2.0f * xb.x;
    ob.y = g * sctx[db + 1] + 2.0f * xb.y;
    ob.z = g * sctx[db + 2] + 2.0f * xb.z;
    ob.w = g * sctx[db + 3] + 2.0f * xb.w;
    float* orow = pout + (size_t)row * DD;
    *(float4*)(orow + da) = oa;
    *(float4*)(orow + db) = ob;
}

extern "C" void kernel_launch(void* const* d_in, const int* in_sizes, int n_in,
                              void* d_out, int out_size, void* d_ws, size_t ws_size,
                              hipStream_t stream) {
    const float* x        = (const float*)d_in[0];
    const float* p        = (const float*)d_in[1];
    const int*   mask     = (const int*)d_in[2];
    const float* ra1_Wk   = (const float*)d_in[5];
    const float* ra1_Wv   = (const float*)d_in[7];
    const float* ra1_bv   = (const float*)d_in[8];
    const float* ra1_ws_k = (const float*)d_in[10];
    const float* ra1_wg_o = (const float*)d_in[12];
    const float* ra1_wg_p = (const float*)d_in[13];
    const float* ra1_bg   = (const float*)d_in[14];
    const float* ra2_Wk   = (const float*)d_in[17];
    const float* ra2_Wv   = (const float*)d_in[19];
    const float* ra2_bv   = (const float*)d_in[20];
    const float* ra2_ws_k = (const float*)d_in[22];
    const float* ra2_wg_o = (const float*)d_in[24];
    const float* ra2_wg_p = (const float*)d_in[25];
    const float* ra2_bg   = (const float*)d_in[26];

    float* ws     = (float*)d_ws;
    float* u1     = ws;                    // 256
    float* u2     = ws + 256;              // 256
    float* wsum1  = ws + 512;              // 256
    float* wsum2  = ws + 768;              // 256
    float* pbar   = ws + 1024;             // 16*256
    float* ctx1   = pbar + BB * DD;        // 16*256
    float* xbar   = ctx1 + BB * DD;        // 16*256
    float* ctx2   = xbar + BB * DD;        // 16*256
    float* gbias1 = ctx2 + BB * DD;        // 16
    float* gbias2 = gbias1 + BB;           // 16
    float* smax   = gbias2 + BB;           // 16
    float* ssum   = smax + BB;             // 16
    float* sk2    = ssum + BB;             // 16*4096

    float* x_out = (float*)d_out;
    float* p_out = x_out + (size_t)BB * SS * DD;

    prep_vectors<<<dim3(2), dim3(DD), 0, stream>>>(ra1_Wk, ra1_ws_k, ra1_wg_o, ra1_wg_p,
                                                   ra2_Wk, ra2_ws_k, ra2_wg_o, ra2_wg_p,
                                                   u1, u2, wsum1, wsum2);
    pass1_attn<<<dim3(BB), dim3(DD), 0, stream>>>(p, u1, pbar);
    wmma_gemm16<<<dim3(DD / 16), dim3(32), 0, stream>>>(pbar, ra1_Wv, ra1_bv, ctx1);
    gate_bias<<<dim3(BB), dim3(DD), 0, stream>>>(ctx1, ra1_wg_o, ra1_bg, gbias1);
    fused_x_kernel<<<dim3(BB * SS / 8), dim3(DD), 0, stream>>>(x, ctx1, wsum1, u2, gbias1,
                                                               mask, x_out, sk2);
    softmax_stats<<<dim3(BB), dim3(DD), 0, stream>>>(sk2, smax, ssum);
    xbar_kernel<<<dim3(BB), dim3(DD), 0, stream>>>(x_out, sk2, smax, ssum, xbar);
    wmma_gemm16<<<dim3(DD / 16), dim3(32), 0, stream>>>(xbar, ra2_Wv, ra2_bv, ctx2);
    gate_bias<<<dim3(BB), dim3(DD), 0, stream>>>(ctx2, ra2_wg_o, ra2_bg, gbias2);
    fused_p_kernel<<<dim3(BB * RR / 8), dim3(DD), 0, stream>>>(p, ctx2, wsum2, gbias2, p_out);
}